// CustomLoss_26800595927517
// MI455X (gfx1250) — compile-verified
//
#include <hip/hip_runtime.h>

// SATD loss on MI455X (gfx1250, wave32).
// diff = orig - pred per 8x8 block; T = H8 * diff; out = sum|T|.
// Memory-bound: ~805 MB @ 23.3 TB/s => ~35us floor. f32 WMMA keeps exactness.
//
// Mapping: 4 consecutive 8x8 blocks -> one 16x16 WMMA tile.
//   A = blockdiag(H8, H8)  (constant, per-lane computed)
//   B = [[D0, D1], [D2, D3]]
//   A*B = [[H*D0, H*D1], [H*D2, H*D3]]  -> all 256 outputs useful.
// K=16 realized as 4 chained V_WMMA_F32_16X16X4_F32.

typedef __attribute__((ext_vector_type(2))) float v2f;
typedef __attribute__((ext_vector_type(8))) float v8f;

#define WG_SIZE 256
#define WAVES_PER_WG (WG_SIZE / 32)
#define NUM_WG 2048

__global__ __launch_bounds__(WG_SIZE) void satd_wmma_kernel(
    const float* __restrict__ orig, const float* __restrict__ pred,
    float* __restrict__ partial, long totalTiles) {
  const int lane = threadIdx.x & 31;
  const int waveInWg = threadIdx.x >> 5;
  const long globalWave = (long)blockIdx.x * WAVES_PER_WG + waveInWg;
  const long numWaves = (long)gridDim.x * WAVES_PER_WG;

  const int mn = lane & 15;          // M for A, N for B
  const int koff = (lane >> 4) << 1; // lanes 16-31 hold K+2/K+3 per ISA layout

  // A = blockdiag(H8,H8); H8[i][j] = (-1)^popcount(i&j) (Sylvester, symmetric).
  // Per step s, VGPR v: K = 4s + v + koff, M = mn.
  v2f a[4];
#pragma unroll
  for (int s = 0; s < 4; ++s) {
#pragma unroll
    for (int v = 0; v < 2; ++v) {
      int k = 4 * s + v + koff;
      float val = 0.0f;
      if ((mn >> 3) == (k >> 3))  // inside one of the two diagonal H8 blocks
        val = (__popc((mn & 7) & (k & 7)) & 1) ? -1.0f : 1.0f;
      a[s][v] = val;
    }
  }

  float acc = 0.0f;

  // Grid-stride over tiles of 4 blocks (256 floats each). Loop bound is wave-
  // uniform, so EXEC stays all-ones across every WMMA (ISA requirement).
  for (long tile = globalWave; tile < totalTiles; tile += numWaves) {
    const long base = tile * 256;

    v8f c = {};  // f32 16x16 accumulator, 8 VGPRs
#pragma unroll
    for (int s = 0; s < 4; ++s) {
      v2f b;
#pragma unroll
      for (int v = 0; v < 2; ++v) {
        int k = 4 * s + v + koff;            // row of the 16x16 B tile
        int blk = ((k >> 3) << 1) | (mn >> 3);  // which of the 4 blocks
        long addr = base + (long)blk * 64 + (long)(k & 7) * 8 + (mn & 7);
        b[v] = orig[addr] - pred[addr];
      }
      // D = A*B + C ; emits v_wmma_f32_16x16x4_f32
      c = __builtin_amdgcn_wmma_f32_16x16x4_f32(
          /*neg_a=*/false, a[s], /*neg_b=*/false, b,
          /*c_mod=*/(short)0, c, /*reuse_a=*/false, /*reuse_b=*/false);
    }

    // Every element of the 16x16 result is a valid transform output.
#pragma unroll
    for (int v = 0; v < 8; ++v) acc += fabsf(c[v]);
  }

  // wave32 reduction
#pragma unroll
  for (int off = 16; off > 0; off >>= 1) acc += __shfl_xor(acc, off, 32);

  __shared__ float smem[WAVES_PER_WG];
  if (lane == 0) smem[waveInWg] = acc;
  __syncthreads();
  if (threadIdx.x == 0) {
    float s = 0.0f;
#pragma unroll
    for (int i = 0; i < WAVES_PER_WG; ++i) s += smem[i];
    partial[blockIdx.x] = s;  // fully overwritten every call (deterministic)
  }
}

__global__ __launch_bounds__(1024) void satd_reduce_kernel(
    const float* __restrict__ partial, float* __restrict__ out, int n) {
  float acc = 0.0f;
  for (int i = threadIdx.x; i < n; i += blockDim.x) acc += partial[i];
#pragma unroll
  for (int off = 16; off > 0; off >>= 1) acc += __shfl_xor(acc, off, 32);
  __shared__ float smem[32];
  const int lane = threadIdx.x & 31;
  const int wv = threadIdx.x >> 5;
  if (lane == 0) smem[wv] = acc;
  __syncthreads();
  if (threadIdx.x == 0) {
    float s = 0.0f;
    const int nw = blockDim.x >> 5;
    for (int i = 0; i < nw; ++i) s += smem[i];
    out[0] = s;
  }
}

extern "C" void kernel_launch(void* const* d_in, const int* in_sizes, int n_in,
                              void* d_out, int out_size, void* d_ws, size_t ws_size,
                              hipStream_t stream) {
  const float* orig = (const float*)d_in[0];
  const float* pred = (const float*)d_in[1];
  float* out = (float*)d_out;
  float* partial = (float*)d_ws;  // NUM_WG floats = 8 KB scratch

  // in_sizes[0] = 524288*3*8*8 = 100,663,296 floats; 256 floats per 4-block tile.
  const long totalTiles = (long)in_sizes[0] / 256;

  satd_wmma_kernel<<<NUM_WG, WG_SIZE, 0, stream>>>(orig, pred, partial, totalTiles);
  satd_reduce_kernel<<<1, 1024, 0, stream>>>(partial, out, NUM_WG);
}